// Val2Cst_Layer_9191230013855
// MI455X (gfx1250) — compile-verified
//
#include <hip/hip_runtime.h>
#include <hip/hip_bf16.h>

typedef float v2f __attribute__((ext_vector_type(2)));
typedef float v8f __attribute__((ext_vector_type(8)));

#define H   128
#define H2  256

__device__ __forceinline__ v8f wmma_f32_16x16x4(v2f a, v2f b, v8f c) {
  // 8 args: (neg_a, A, neg_b, B, c_mod, C, reuse_a, reuse_b)
  return __builtin_amdgcn_wmma_f32_16x16x4_f32(false, a, false, b, (short)0, c,
                                               false, false);
}

// ---------------------------------------------------------------------------
// Kernel 1: fused per-node MLP.
//   x1    = relu([h_val, assign] @ W1 + b1)          [16 x 128] per block
//   x_val = LayerNorm(x1 @ W2; g1, beta1)            -> d_out (x_val part)
//   m_val = LayerNorm(x_val @ W3; g2, beta2)[2N,128] -> workspace
// One block = 16 nodes, 8 waves; each wave owns a 16-column WMMA tile.
// ---------------------------------------------------------------------------
__global__ __launch_bounds__(256) void fused_mlp_kernel(
    const float* __restrict__ h_val, const float* __restrict__ assign,
    const float* __restrict__ W1, const float* __restrict__ b1,
    const float* __restrict__ W2, const float* __restrict__ g1,
    const float* __restrict__ beta1, const float* __restrict__ W3,
    const float* __restrict__ g2, const float* __restrict__ beta2,
    float* __restrict__ xval_out, float* __restrict__ mval, int N)
{
  __shared__ float s_x0[16][132];   // input tile (h_val rows)
  __shared__ float s_x1[16][132];   // relu(x @ W1 + ...)
  __shared__ float s_d2[16][132];   // x1 @ W2 (pre-LN)
  __shared__ float s_xv[16][132];   // x_val (post-LN)
  __shared__ float s_d3[16][264];   // x_val @ W3 (pre-LN)
  __shared__ float s_as[16];        // assign per row
  __shared__ float s_r1[16][17];    // LN partial sums
  __shared__ float s_r2[16][17];    // LN partial sumsq
  __shared__ float s_mu[16];
  __shared__ float s_rs[16];

  const int tid  = threadIdx.x;
  const int wid  = tid >> 5;        // wave 0..7
  const int lane = tid & 31;
  const int ml   = lane & 15;       // M index within A/B tile
  const int sh   = lane >> 4;       // K sub-half (0/1)
  const int base = blockIdx.x * 16;
  const int nrow = min(16, N - base);

  // --- cooperative load of 16x128 input tile + assign -----------------------
  #pragma unroll
  for (int i = 0; i < 8; ++i) {
    int idx = i * 256 + tid;
    int r = idx >> 7, c = idx & 127;
    int rr = min(base + r, N - 1);
    s_x0[r][c] = h_val[(size_t)rr * H + c];
  }
  if (tid < 16) s_as[tid] = assign[min(base + tid, N - 1)];
  __syncthreads();

  // --- GEMM1: x1 = relu(x0 @ W1[0:128] + assign*W1[128] + b1) ---------------
  {
    const int n0 = wid * 16;
    v8f acc = {0.f, 0.f, 0.f, 0.f, 0.f, 0.f, 0.f, 0.f};
    #pragma unroll
    for (int kk = 0; kk < 32; ++kk) {
      const int k = kk * 4 + 2 * sh;
      v2f a; a.x = s_x0[ml][k]; a.y = s_x0[ml][k + 1];
      const float* wp = W1 + (size_t)k * H + n0 + ml;
      v2f b; b.x = wp[0]; b.y = wp[H];
      acc = wmma_f32_16x16x4(a, b, acc);
    }
    const int   col = n0 + ml;
    const float wl  = W1[(size_t)H * H + col];  // last row (assign column)
    const float bb  = b1[col];
    #pragma unroll
    for (int i = 0; i < 8; ++i) {
      const int m = i + 8 * sh;
      float v = acc[i] + s_as[m] * wl + bb;
      s_x1[m][col] = fmaxf(v, 0.f);
    }
  }
  __syncthreads();

  // --- GEMM2: d2 = x1 @ W2 --------------------------------------------------
  {
    const int n0 = wid * 16;
    v8f acc = {0.f, 0.f, 0.f, 0.f, 0.f, 0.f, 0.f, 0.f};
    #pragma unroll
    for (int kk = 0; kk < 32; ++kk) {
      const int k = kk * 4 + 2 * sh;
      v2f a; a.x = s_x1[ml][k]; a.y = s_x1[ml][k + 1];
      const float* wp = W2 + (size_t)k * H + n0 + ml;
      v2f b; b.x = wp[0]; b.y = wp[H];
      acc = wmma_f32_16x16x4(a, b, acc);
    }
    const int col = n0 + ml;
    #pragma unroll
    for (int i = 0; i < 8; ++i) s_d2[i + 8 * sh][col] = acc[i];
  }
  __syncthreads();

  // --- LayerNorm over 128 cols -> x_val (LDS + global out) ------------------
  {
    const int r = tid >> 4, j = tid & 15;
    float sm = 0.f, sq = 0.f;
    #pragma unroll
    for (int c = j; c < H; c += 16) { float v = s_d2[r][c]; sm += v; sq += v * v; }
    s_r1[r][j] = sm; s_r2[r][j] = sq;
    __syncthreads();
    if (tid < 16) {
      float a = 0.f, b = 0.f;
      #pragma unroll
      for (int j2 = 0; j2 < 16; ++j2) { a += s_r1[tid][j2]; b += s_r2[tid][j2]; }
      float mu  = a * (1.0f / H);
      float var = b * (1.0f / H) - mu * mu;
      s_mu[tid] = mu;
      s_rs[tid] = rsqrtf(var + 1e-5f);
    }
    __syncthreads();
    const float mu = s_mu[r], rs = s_rs[r];
    #pragma unroll
    for (int c = j; c < H; c += 16) {
      float v = (s_d2[r][c] - mu) * rs * g1[c] + beta1[c];
      s_xv[r][c] = v;
      if (r < nrow) xval_out[(size_t)(base + r) * H + c] = v;
    }
  }
  __syncthreads();

  // --- GEMM3: d3 = x_val @ W3 (16x256; each wave does 2 column tiles) -------
  {
    const int n0a = wid * 32, n0b = n0a + 16;
    v8f acc0 = {0.f, 0.f, 0.f, 0.f, 0.f, 0.f, 0.f, 0.f};
    v8f acc1 = {0.f, 0.f, 0.f, 0.f, 0.f, 0.f, 0.f, 0.f};
    #pragma unroll
    for (int kk = 0; kk < 32; ++kk) {
      const int k = kk * 4 + 2 * sh;
      v2f a; a.x = s_xv[ml][k]; a.y = s_xv[ml][k + 1];
      const float* wp = W3 + (size_t)k * H2;
      v2f b0; b0.x = wp[n0a + ml]; b0.y = wp[H2 + n0a + ml];
      v2f b1v; b1v.x = wp[n0b + ml]; b1v.y = wp[H2 + n0b + ml];
      acc0 = wmma_f32_16x16x4(a, b0, acc0);
      acc1 = wmma_f32_16x16x4(a, b1v, acc1);
    }
    #pragma unroll
    for (int i = 0; i < 8; ++i) {
      const int m = i + 8 * sh;
      s_d3[m][n0a + ml] = acc0[i];
      s_d3[m][n0b + ml] = acc1[i];
    }
  }
  __syncthreads();

  // --- LayerNorm over 256 cols -> m_val rows [2n, 2n+1] ---------------------
  {
    const int r = tid >> 4, j = tid & 15;
    float sm = 0.f, sq = 0.f;
    #pragma unroll
    for (int c = j; c < H2; c += 16) { float v = s_d3[r][c]; sm += v; sq += v * v; }
    s_r1[r][j] = sm; s_r2[r][j] = sq;
    __syncthreads();
    if (tid < 16) {
      float a = 0.f, b = 0.f;
      #pragma unroll
      for (int j2 = 0; j2 < 16; ++j2) { a += s_r1[tid][j2]; b += s_r2[tid][j2]; }
      float mu  = a * (1.0f / H2);
      float var = b * (1.0f / H2) - mu * mu;
      s_mu[tid] = mu;
      s_rs[tid] = rsqrtf(var + 1e-5f);
    }
    __syncthreads();
    const float mu = s_mu[r], rs = s_rs[r];
    if (r < nrow) {
      #pragma unroll
      for (int c = j; c < H2; c += 16) {
        float v = (s_d3[r][c] - mu) * rs * g2[c] + beta2[c];
        mval[(size_t)(2 * (base + r) + (c >> 7)) * H + (c & 127)] = v;
      }
    }
  }
}

// ---------------------------------------------------------------------------
// Kernel 2: zero r_cst
// ---------------------------------------------------------------------------
__global__ void zero_kernel(float4* __restrict__ p, int n4) {
  int i = blockIdx.x * blockDim.x + threadIdx.x;
  if (i < n4) p[i] = make_float4(0.f, 0.f, 0.f, 0.f);
}

// ---------------------------------------------------------------------------
// Kernel 3: edge gather + atomic scatter-sum into constraints.
// One wave per edge; each lane handles 4 of the 128 columns.
// ---------------------------------------------------------------------------
__global__ __launch_bounds__(256) void scatter_kernel(
    const int* __restrict__ ecst, const int* __restrict__ eval,
    const int* __restrict__ ele, const float* __restrict__ mval,
    float* __restrict__ rcst, int E)
{
  const int lane = threadIdx.x & 31;
  const int gw   = (int)((blockIdx.x * blockDim.x + threadIdx.x) >> 5);
  const int nw   = (int)((gridDim.x * blockDim.x) >> 5);
  for (int e = gw; e < E; e += nw) {
    const int c   = ecst[e];
    const int src = 2 * eval[e] + ele[e];
    const float4 v = *(const float4*)(mval + (size_t)src * H + lane * 4);
    float* dst = rcst + (size_t)c * H + lane * 4;
    __hip_atomic_fetch_add(dst + 0, v.x, __ATOMIC_RELAXED, __HIP_MEMORY_SCOPE_AGENT);
    __hip_atomic_fetch_add(dst + 1, v.y, __ATOMIC_RELAXED, __HIP_MEMORY_SCOPE_AGENT);
    __hip_atomic_fetch_add(dst + 2, v.z, __ATOMIC_RELAXED, __HIP_MEMORY_SCOPE_AGENT);
    __hip_atomic_fetch_add(dst + 3, v.w, __ATOMIC_RELAXED, __HIP_MEMORY_SCOPE_AGENT);
  }
}

// ---------------------------------------------------------------------------
extern "C" void kernel_launch(void* const* d_in, const int* in_sizes, int n_in,
                              void* d_out, int out_size, void* d_ws, size_t ws_size,
                              hipStream_t stream)
{
  const float* h_val  = (const float*)d_in[0];
  const float* assign = (const float*)d_in[1];
  const int*   ecst   = (const int*)d_in[2];
  const int*   eval   = (const int*)d_in[3];
  const int*   ele    = (const int*)d_in[4];
  // d_in[5] = num_cst (device scalar, derived from out_size instead)
  const float* W1     = (const float*)d_in[6];
  const float* b1     = (const float*)d_in[7];
  const float* W2     = (const float*)d_in[8];
  const float* g1     = (const float*)d_in[9];
  const float* be1    = (const float*)d_in[10];
  const float* W3     = (const float*)d_in[11];
  const float* g2     = (const float*)d_in[12];
  const float* be2    = (const float*)d_in[13];

  const int N  = in_sizes[1];          // NUM_VAL
  const int E  = in_sizes[2];          // edges
  const int NC = out_size / H - N;     // NUM_CST

  float* rcst = (float*)d_out;                         // [NC, 128]
  float* xval = (float*)d_out + (size_t)NC * H;        // [N, 128]
  float* mval = (float*)d_ws;                          // [2N, 128]

  {
    int n4 = (NC * H) / 4;
    zero_kernel<<<(n4 + 255) / 256, 256, 0, stream>>>((float4*)rcst, n4);
  }
  fused_mlp_kernel<<<(N + 15) / 16, 256, 0, stream>>>(
      h_val, assign, W1, b1, W2, g1, be1, W3, g2, be2, xval, mval, N);
  scatter_kernel<<<2048, 256, 0, stream>>>(ecst, eval, ele, mval, rcst, E);

  (void)n_in; (void)ws_size;
}